// YoloLoss_1872605741482
// MI455X (gfx1250) — compile-verified
//
#include <hip/hip_runtime.h>

typedef float v2f __attribute__((ext_vector_type(2)));
typedef float v8f __attribute__((ext_vector_type(8)));

#define YS 7
#define CH 30
#define NBLOCKS 1024
#define TPB 256
#define NWAVES (TPB / 32)

// ---------------------------------------------------------------------------
// Wave32 xor-shuffle sum (5 steps -> every lane holds the full wave sum).
// ---------------------------------------------------------------------------
__device__ __forceinline__ float wave_sum(float v) {
    v += __shfl_xor(v, 1, 32);
    v += __shfl_xor(v, 2, 32);
    v += __shfl_xor(v, 4, 32);
    v += __shfl_xor(v, 8, 32);
    v += __shfl_xor(v, 16, 32);
    return v;
}

// ---------------------------------------------------------------------------
// Block reduction of 5 accumulators using V_WMMA_F32_16X16X4_F32 as a
// full-precision row-sum engine:  D = A x ones  =>  D[m][*] = sum_k A[m][k].
// A layout (ISA 7.12.2, 32-bit A 16x4): lanes 0-15 hold K=0 (vgpr0) / K=1
// (vgpr1); lanes 16-31 hold K=2 / K=3.  Rows m=0..4 carry the 5 loss terms.
// Thread 0 stores the 5 scaled totals to dst[0..4].
// ---------------------------------------------------------------------------
__device__ __forceinline__ void block_reduce_store(float acc[5], float* dst,
                                                   float s0, float s1, float s2,
                                                   float s3, float s4) {
    __shared__ float W[NWAVES][5];
    const int lane = threadIdx.x & 31;
    const int wave = threadIdx.x >> 5;

#pragma unroll
    for (int c = 0; c < 5; ++c) acc[c] = wave_sum(acc[c]);

    if (lane == 0) {
#pragma unroll
        for (int c = 0; c < 5; ++c) W[wave][c] = acc[c];
    }
    __syncthreads();

    // Pre-fold 8 wave partials into the 4 K-slots of row m (m = component).
    const int m = lane & 15;
    float a0 = 0.0f, a1 = 0.0f;
    if (m < 5) {
        if (lane < 16) { a0 = W[0][m] + W[1][m]; a1 = W[2][m] + W[3][m]; }
        else           { a0 = W[4][m] + W[5][m]; a1 = W[6][m] + W[7][m]; }
    }
    v2f A;  A.x = a0;   A.y = a1;
    v2f Bv; Bv.x = 1.0f; Bv.y = 1.0f;     // ones matrix: layout-independent
    v8f C = {};
    // (neg_a, A, neg_b, B, c_mod, C, reuse_a, reuse_b)
    v8f D = __builtin_amdgcn_wmma_f32_16x16x4_f32(false, A, false, Bv,
                                                  (short)0, C, false, false);
    // D vgpr p, lanes 0-15 = row M=p -> component totals live in D[0..4], lane 0.
    if (threadIdx.x == 0) {
        dst[0] = D[0] * s0;
        dst[1] = D[1] * s1;
        dst[2] = D[2] * s2;
        dst[3] = D[3] * s3;
        dst[4] = D[4] * s4;
    }
}

// ---------------------------------------------------------------------------
// IoU exactly as the reference (_convert_box + _iou).
// ---------------------------------------------------------------------------
__device__ __forceinline__ float iou_f(float ax, float ay, float aw, float ah,
                                       float bx, float by, float bw, float bh,
                                       float gi, float gj) {
    const float step = 1.0f / 7.0f;
    float x1 = (ax + gj) * step, y1 = (ay + gi) * step;
    float x2 = (bx + gj) * step, y2 = (by + gi) * step;
    float xm1 = fmaxf(x1 - 0.5f * aw, 0.0f), ym1 = fmaxf(y1 - 0.5f * ah, 0.0f);
    float xm2 = fmaxf(x2 - 0.5f * bw, 0.0f), ym2 = fmaxf(y2 - 0.5f * bh, 0.0f);
    float iw = aw + bw - (fmaxf(xm1 + aw, xm2 + bw) - fminf(xm1, xm2));
    float ih = ah + bh - (fmaxf(ym1 + ah, ym2 + bh) - fminf(ym1, ym2));
    float inter = iw * ih;
    float uni   = aw * ah + bw * bh - inter;
    return (iw > 0.0f && ih > 0.0f) ? inter / uni : 0.0f;
}

// ---------------------------------------------------------------------------
// Kernel 1: streaming pass. One cell per lane per iteration, grid-stride.
// Emits per-block partials (5 floats) to d_ws.
// ---------------------------------------------------------------------------
__global__ void __launch_bounds__(TPB)
yolo_partial(const float* __restrict__ pred, const float* __restrict__ target,
             float* __restrict__ partial, int ncells) {
    float acc[5] = {0.f, 0.f, 0.f, 0.f, 0.f};
    const int stride = gridDim.x * blockDim.x;

    for (int cell = blockIdx.x * blockDim.x + threadIdx.x; cell < ncells;
         cell += stride) {
        const float2* p2 = (const float2*)(pred   + (size_t)cell * CH);
        const float2* t2 = (const float2*)(target + (size_t)cell * CH);
        float pv[CH], tv[CH];
#pragma unroll
        for (int k = 0; k < CH / 2; ++k) {
            float2 a = p2[k]; pv[2 * k] = a.x; pv[2 * k + 1] = a.y;
            float2 b = t2[k]; tv[2 * k] = b.x; tv[2 * k + 1] = b.y;
        }

        const int j = cell % YS;
        const int i = (cell / YS) % YS;
        const float gj = (float)j, gi = (float)i;

        const float iou1 = iou_f(tv[0], tv[1], tv[2], tv[3],
                                 pv[0], pv[1], pv[2], pv[3], gi, gj);
        const float iou2 = iou_f(tv[0], tv[1], tv[2], tv[3],
                                 pv[5], pv[6], pv[7], pv[8], gi, gj);
        const bool  resp1  = iou1 > iou2;
        const float tconf1 = resp1 ? iou1 : 0.0f;
        const float tconf2 = resp1 ? 0.0f : iou2;
        const float posf   = (tv[4] > 0.0f)  ? 1.0f : 0.0f;
        const float noobjf = (tv[4] == 0.0f) ? 1.0f : 0.0f;

        // obj
        float d1 = tconf1 - pv[4], d2 = tconf2 - pv[9];
        acc[0] += posf * (d1 * d1 + d2 * d2);
        // noobj
        float n1 = tv[4] - pv[4], n2 = tv[9] - pv[9];
        acc[1] += noobjf * (n1 * n1 + n2 * n2);
        // xy
        float e0 = tv[0] - pv[0], e1 = tv[1] - pv[1];
        float e5 = tv[5] - pv[5], e6 = tv[6] - pv[6];
        acc[2] += posf * (e0 * e0 + e1 * e1 + e5 * e5 + e6 * e6);
        // wh (sqrt space)
        float w2 = sqrtf(tv[2]) - sqrtf(pv[2]);
        float w3 = sqrtf(tv[3]) - sqrtf(pv[3]);
        float w7 = sqrtf(tv[7]) - sqrtf(pv[7]);
        float w8 = sqrtf(tv[8]) - sqrtf(pv[8]);
        acc[3] += posf * (w2 * w2 + w3 * w3 + w7 * w7 + w8 * w8);
        // class
        float cls = 0.0f;
#pragma unroll
        for (int c = 10; c < CH; ++c) {
            float d = tv[c] - pv[c];
            cls += d * d;
        }
        acc[4] += posf * cls;
    }

    block_reduce_store(acc, partial + (size_t)blockIdx.x * 5,
                       1.f, 1.f, 1.f, 1.f, 1.f);
}

// ---------------------------------------------------------------------------
// Kernel 2: single block folds NBLOCKS x 5 partials, applies lambda / bs.
// ---------------------------------------------------------------------------
__global__ void __launch_bounds__(TPB)
yolo_final(const float* __restrict__ partial, float* __restrict__ out,
           int nparts) {
    float acc[5] = {0.f, 0.f, 0.f, 0.f, 0.f};
    for (int b = threadIdx.x; b < nparts; b += blockDim.x) {
#pragma unroll
        for (int c = 0; c < 5; ++c) acc[c] += partial[(size_t)b * 5 + c];
    }
    const float inv_bs = 1.0f / 16384.0f;
    block_reduce_store(acc, out,
                       inv_bs,            // obj_loss / bs
                       0.5f * inv_bs,     // LAMBDA_NOOBJ * noobj / bs
                       5.0f * inv_bs,     // LAMBDA_COORD * xy / bs
                       5.0f * inv_bs,     // LAMBDA_COORD * wh / bs
                       inv_bs);           // class / bs
}

extern "C" void kernel_launch(void* const* d_in, const int* in_sizes, int n_in,
                              void* d_out, int out_size, void* d_ws, size_t ws_size,
                              hipStream_t stream) {
    const float* pred   = (const float*)d_in[0];
    const float* target = (const float*)d_in[1];
    float* out     = (float*)d_out;
    float* partial = (float*)d_ws;                  // NBLOCKS * 5 floats = 20 KB

    const int ncells = in_sizes[0] / CH;            // 16384 * 49 = 802816

    yolo_partial<<<NBLOCKS, TPB, 0, stream>>>(pred, target, partial, ncells);
    yolo_final<<<1, TPB, 0, stream>>>(partial, out, NBLOCKS);
}